// PointPillarVoxelization_19859928776850
// MI455X (gfx1250) — compile-verified
//
#include <hip/hip_runtime.h>
#include <hip/hip_bf16.h>
#include <stdint.h>

// ---------------- problem constants (match reference) ----------------
#define VX 0.16f
#define VY 0.16f
#define VZ 4.0f
#define X0 0.0f
#define Y0 (-39.68f)
#define Z0 (-3.0f)
#define NXC 432
#define NYC 496
#define NZC 1
#define GRIDG (NXC * NYC * NZC)   // 214272 (divisible by 4)
#define MAXP 100
#define MAXV 12000
#define SCAN_B 1024
#define NBLK ((GRIDG + SCAN_B - 1) / SCAN_B)   // 210

// voxel bin exactly as the reference: f32 sub, f32 divide, floor, cast to i32
__device__ __forceinline__ int voxel_lin(float x, float y, float z, bool& valid) {
    float fx = floorf((x - X0) / VX);
    float fy = floorf((y - Y0) / VY);
    float fz = floorf((z - Z0) / VZ);
    int ix = (int)fx, iy = (int)fy, iz = (int)fz;
    valid = (ix >= 0) && (ix < NXC) && (iy >= 0) && (iy < NYC) && (iz >= 0) && (iz < NZC);
    return iz * (NYC * NXC) + iy * NXC + ix;
}

// K0: zero/initialize everything that must be fresh per launch.
// All regions are multiples of 4 elements -> 128-bit stores (global_store_b128).
__global__ void pp_init(float* __restrict__ out, int* __restrict__ counts,
                        int* __restrict__ scnt) {
    int64_t i = blockIdx.x * (int64_t)blockDim.x + threadIdx.x;
    int64_t stride = (int64_t)gridDim.x * blockDim.x;

    const int64_t nvox4   = (int64_t)MAXV * MAXP;      // 1,200,000 float4 (voxel data)
    const int64_t ncoord4 = (int64_t)MAXV * 3 / 4;     // 9,000 float4 (coords = -1)
    const int64_t nnum4   = MAXV / 4;                  // 3,000 float4 (num_points = 0)
    const int64_t ncnt4   = GRIDG / 4;                 // 53,568 int4 (counts = 0)
    const int64_t nscnt4  = MAXV / 4;                  // 3,000 int4 (scnt = 0)

    float4* vox = (float4*)out;
    float4* crd = (float4*)(out + (int64_t)MAXV * MAXP * 4);
    float4* nmp = (float4*)(out + (int64_t)MAXV * MAXP * 4 + (int64_t)MAXV * 3);
    int4*   cnt = (int4*)counts;
    int4*   scn = (int4*)scnt;

    const float4 f0   = make_float4(0.0f, 0.0f, 0.0f, 0.0f);
    const float4 fm1  = make_float4(-1.0f, -1.0f, -1.0f, -1.0f);
    const int4   i0   = make_int4(0, 0, 0, 0);

    for (int64_t t = i; t < nvox4;   t += stride) vox[t] = f0;
    for (int64_t t = i; t < ncoord4; t += stride) crd[t] = fm1;
    for (int64_t t = i; t < nnum4;   t += stride) nmp[t] = f0;
    for (int64_t t = i; t < ncnt4;   t += stride) cnt[t] = i0;
    for (int64_t t = i; t < nscnt4;  t += stride) scn[t] = i0;
}

// K1: per-voxel point counts (order-independent sum -> deterministic).
__global__ void pp_count(const float4* __restrict__ pts, int n, int* __restrict__ counts) {
    int i = blockIdx.x * blockDim.x + threadIdx.x;
    if (i >= n) return;
    if (i + 4096 < n) __builtin_prefetch(&pts[i + 4096], 0, 1);  // global_prefetch_b8
    float4 p = pts[i];
    bool v;
    int lin = voxel_lin(p.x, p.y, p.z, v);
    if (v) atomicAdd(&counts[lin], 1);
}

// K2: per-block reduction of occupancy flags (LDS tree reduce).
__global__ void pp_partial(const int* __restrict__ counts, int* __restrict__ partials) {
    __shared__ int sm[SCAN_B];
    int v = blockIdx.x * SCAN_B + threadIdx.x;
    int f = (v < GRIDG && counts[v] > 0) ? 1 : 0;
    sm[threadIdx.x] = f;
    __syncthreads();
    for (int s = SCAN_B / 2; s > 0; s >>= 1) {
        if (threadIdx.x < s) sm[threadIdx.x] += sm[threadIdx.x + s];
        __syncthreads();
    }
    if (threadIdx.x == 0) partials[blockIdx.x] = sm[0];
}

// K3: exclusive scan of the 210 block partials (tiny, serial, deterministic).
__global__ void pp_scan_partials(int* __restrict__ partials) {
    if (threadIdx.x == 0 && blockIdx.x == 0) {
        int acc = 0;
        for (int b = 0; b < NBLK; ++b) {
            int t = partials[b];
            partials[b] = acc;
            acc += t;
        }
    }
}

// K4: block-local Hillis-Steele scan of flags + partial offsets -> output row
// per occupied voxel (first MAXV occupied voxels in ascending index order).
// Also writes coords and (uncapped, like the reference) num_points.
__global__ void pp_assign(const int* __restrict__ counts, const int* __restrict__ partials,
                          int* __restrict__ rowmap, float* __restrict__ out) {
    __shared__ int sm[SCAN_B];
    int tid = threadIdx.x;
    int v = blockIdx.x * SCAN_B + tid;
    int c = (v < GRIDG) ? counts[v] : 0;
    int f = (c > 0) ? 1 : 0;
    sm[tid] = f;
    __syncthreads();
    for (int off = 1; off < SCAN_B; off <<= 1) {
        int val = (tid >= off) ? sm[tid - off] : 0;
        __syncthreads();
        sm[tid] += val;
        __syncthreads();
    }
    if (v >= GRIDG) return;
    int pos = partials[blockIdx.x] + sm[tid] - f;  // exclusive prefix
    int row = (f && pos < MAXV) ? pos : -1;
    rowmap[v] = row;
    if (row >= 0) {
        int z = v / (NYC * NXC);
        int rem = v % (NYC * NXC);
        int y = rem / NXC;
        int x = rem % NXC;
        float* coords = out + (int64_t)MAXV * MAXP * 4;
        coords[row * 3 + 0] = (float)z;
        coords[row * 3 + 1] = (float)y;
        coords[row * 3 + 2] = (float)x;
        float* nump = coords + (int64_t)MAXV * 3;
        nump[row] = (float)c;
    }
}

// K5: stage points into their row with an atomic ticket + original index.
// The staged SET per row is schedule-independent (counts << MAXP), so the
// subsequent index sort makes the result fully deterministic.
__global__ void pp_stage(const float4* __restrict__ pts, int n,
                         const int* __restrict__ rowmap, int* __restrict__ scnt,
                         int* __restrict__ sidx, float* __restrict__ out) {
    int i = blockIdx.x * blockDim.x + threadIdx.x;
    if (i >= n) return;
    if (i + 4096 < n) __builtin_prefetch(&pts[i + 4096], 0, 1);
    float4 p = pts[i];
    bool valid;
    int lin = voxel_lin(p.x, p.y, p.z, valid);
    if (!valid) return;
    int row = rowmap[lin];
    if (row < 0) return;
    int s = atomicAdd(&scnt[row], 1);
    if (s >= MAXP) return;   // reference drops slots >= 100
    sidx[row * MAXP + s] = i;
    float4* vox = (float4*)out;
    vox[row * MAXP + s] = p;
}

// K6: per-row insertion sort by original point index -> stable slot order
// identical to the reference's stable argsort.
__global__ void pp_sort(const int* __restrict__ scnt, int* __restrict__ sidx,
                        float* __restrict__ out) {
    int r = blockIdx.x * blockDim.x + threadIdx.x;
    if (r >= MAXV) return;
    int k = scnt[r];
    if (k > MAXP) k = MAXP;
    if (k <= 1) return;
    int* idx = sidx + r * MAXP;
    float4* vox = (float4*)out + (int64_t)r * MAXP;
    for (int a = 1; a < k; ++a) {
        int ia = idx[a];
        float4 pa = vox[a];
        int b = a - 1;
        while (b >= 0 && idx[b] > ia) {
            idx[b + 1] = idx[b];
            vox[b + 1] = vox[b];
            --b;
        }
        idx[b + 1] = ia;
        vox[b + 1] = pa;
    }
}

extern "C" void kernel_launch(void* const* d_in, const int* in_sizes, int n_in,
                              void* d_out, int out_size, void* d_ws, size_t ws_size,
                              hipStream_t stream) {
    const float4* pts = (const float4*)d_in[0];
    int n = in_sizes[0] / 4;   // (N,4) float32 -> N points
    float* out = (float*)d_out;

    // workspace layout (ints), ~6.6 MB total
    int* counts   = (int*)d_ws;             // GRIDG
    int* rowmap   = counts + GRIDG;         // GRIDG
    int* scnt     = rowmap + GRIDG;         // MAXV
    int* partials = scnt + MAXV;            // NBLK
    int* sidx     = partials + ((NBLK + 3) & ~3);  // MAXV*MAXP

    pp_init<<<2048, 256, 0, stream>>>(out, counts, scnt);
    int pb = (n + 255) / 256;
    pp_count<<<pb, 256, 0, stream>>>(pts, n, counts);
    pp_partial<<<NBLK, SCAN_B, 0, stream>>>(counts, partials);
    pp_scan_partials<<<1, 64, 0, stream>>>(partials);
    pp_assign<<<NBLK, SCAN_B, 0, stream>>>(counts, partials, rowmap, out);
    pp_stage<<<pb, 256, 0, stream>>>(pts, n, rowmap, scnt, sidx, out);
    pp_sort<<<(MAXV + 255) / 256, 256, 0, stream>>>(scnt, sidx, out);
}